// EncoderVMP_51015621542088
// MI455X (gfx1250) — compile-verified
//
// CDNA5 (gfx1250 / MI455X) implementation of the VMP transformer encoder.
//
// Precision: fp32 end-to-end, matrix math on V_WMMA_F32_16X16X4_F32 (the
// native f32 WMMA shape; f16 inputs would wreck the variance path).
// Data movement: attention scores (134MB x2 per layer) are processed in
// (b,h)-chunks sized to stay resident in the 192MB L2 between the
// score-GEMM -> softmax -> AV-GEMM sequence, so they never round-trip HBM.
// Each wave register-blocks a 32x32 output region (2x2 WMMA tiles, mean+var):
// 8 b64 fragment loads feed 12 WMMAs per K=4 step.
//
// Input flattening assumption: depth-first dict *insertion* order of
// setup_inputs(): x, emb_W, emb_vW, emb_b, emb_vb,
// 2x layer dicts (Wq,vWq,Wk,vWk,Wv,vWv,fc_W,fc_vW,fc_b,fc_vb,n2_g,n2_b,n2_vg,n2_vb),
// 2x norm1 dicts (g,b,vg,vb), fin_Wk, fin_vWk, fin_Wv, fin_vWv  -> 45 tensors.

#include <hip/hip_runtime.h>
#include <hip/hip_bf16.h>
#include <math.h>
#include <stdint.h>

#define S_LEN   1024
#define DIN     128
#define DMODEL  512
#define NHEAD   8
#define DHEAD   64
#define NBATCH  4
#define BS      (NBATCH * S_LEN)     /* 4096 rows */
#define BH      (NBATCH * NHEAD)     /* 32 attention batches */
#define LN_TOL  1e-6f

typedef float v2f __attribute__((ext_vector_type(2)));
typedef float v8f __attribute__((ext_vector_type(8)));

#if defined(__has_builtin)
#  if __has_builtin(__builtin_amdgcn_wmma_f32_16x16x4_f32)
#    define HAVE_WMMA_F32 1
#  endif
#endif
#ifndef HAVE_WMMA_F32
#  define HAVE_WMMA_F32 0
#endif

#define WMMA_F32(A, B, C) \
    __builtin_amdgcn_wmma_f32_16x16x4_f32(false, (A), false, (B), (short)0, (C), false, false)

// ---------------------------------------------------------------------------
// Generic VMP GEMM:
//   om[m,n] = scale_m * sum_k xm[m,k]*W[n,k]                        (+bias)(+pos)
//   ov[m,n] = scale_v * sum_k ( vx[m,k]*(W^2+vW)[n,k] + xm^2*vW )   (+vbias)
// A is row-major [M,K] (k-stride 1, row-stride lda).
// B element (n,k) lives at wm[n*ldb_n + k*ldb_k]  (covers W^T and V layouts).
// Template: HAS_VX  - vx term present (false for the embedding, where vx==0)
//           BK1     - ldb_k == 1 (contiguous K in B -> b64 pair loads)
// Batched over blockIdx.y with per-operand (b,h) offset bases, so one kernel
// serves plain linears, QK^T scores, and the A*V contraction.
// Each wave computes a 32x32 output region: 2x2 tiles of 16x16, for both the
// mean and the variance accumulators (8 x v8f accumulators total).
// NOTE: grids must tile (M/32)x(N/32) exactly (no bounds guard -> no EXEC
// games around the WMMAs). All call sites below satisfy this.
// ---------------------------------------------------------------------------
template <bool HAS_VX, bool BK1>
__global__ __launch_bounds__(256) void gemm_vmp_wmma(
    const float* __restrict__ xm, const float* __restrict__ xv,
    const float* __restrict__ wm, const float* __restrict__ wv,
    const float* __restrict__ bias, const float* __restrict__ vbias,
    const float* __restrict__ pos, int posS,
    float* __restrict__ om, float* __restrict__ ov,
    int N, int K,
    int lda, long long ldb_n, long long ldb_k, int ldo,
    int Hdim,
    int x_bh0, long long xs_b, long long xs_h,
    int w_bh0, long long ws_b, long long ws_h,
    int o_bh0, long long os_b, long long os_h,
    float scale_m, float scale_v, int do_relu)
{
    const int wave = threadIdx.x >> 5;
    const int lane = threadIdx.x & 31;
    const int r    = lane & 15;     // row/col-within-tile for this lane
    const int half = lane >> 4;     // 0: lanes 0-15, 1: lanes 16-31

    // 32x32 super-tiles
    const int       tilesN2 = N >> 5;
    const long long tile    = (long long)blockIdx.x * (blockDim.x >> 5) + wave;
    const int tm32 = (int)(tile / tilesN2) * 32;
    const int tn32 = (int)(tile % tilesN2) * 32;

    const int bx = x_bh0 + (int)blockIdx.y;
    const int bw = w_bh0 + (int)blockIdx.y;
    const int bo = o_bh0 + (int)blockIdx.y;
    const long long xoff = (long long)(bx / Hdim) * xs_b + (long long)(bx % Hdim) * xs_h;
    const long long woff = (long long)(bw / Hdim) * ws_b + (long long)(bw % Hdim) * ws_h;
    const long long ooff = (long long)(bo / Hdim) * os_b + (long long)(bo % Hdim) * os_h;

    const long long bk = BK1 ? 1LL : ldb_k;   // compile-time 1 in the BK1 case

    // Walking pointers for the two A row-tiles and two B col-tiles.
    const float* xp0  = xm + xoff + (long long)(tm32 + r) * lda + 2 * half;
    const float* xp1  = xp0 + 16LL * lda;
    const float* vxp0 = HAS_VX ? (xv + xoff + (long long)(tm32 + r) * lda + 2 * half)
                               : (const float*)0;
    const float* vxp1 = HAS_VX ? (vxp0 + 16LL * lda) : (const float*)0;
    const float* wp0  = wm + woff + (long long)(tn32 + r) * ldb_n + (long long)(2 * half) * bk;
    const float* wp1  = wp0 + 16LL * ldb_n;
    const float* vwp0 = wv + woff + (long long)(tn32 + r) * ldb_n + (long long)(2 * half) * bk;
    const float* vwp1 = vwp0 + 16LL * ldb_n;

    v8f cm00 = {0.f,0.f,0.f,0.f,0.f,0.f,0.f,0.f}, cm01 = cm00, cm10 = cm00, cm11 = cm00;
    v8f cv00 = cm00, cv01 = cm00, cv10 = cm00, cv11 = cm00;

    __builtin_prefetch(xp0, 0, 1);              // -> global_prefetch_b8
    __builtin_prefetch(wp0, 0, 1);

#if HAVE_WMMA_F32
#pragma unroll 2
    for (int k0 = 0; k0 < K; k0 += 4) {
        v2f ax0;  ax0.x  = xp0[0];  ax0.y  = xp0[1];
        v2f ax1;  ax1.x  = xp1[0];  ax1.y  = xp1[1];
        v2f bw0;  bw0.x  = wp0[0];  bw0.y  = wp0[bk];
        v2f bw1;  bw1.x  = wp1[0];  bw1.y  = wp1[bk];
        v2f bvw0; bvw0.x = vwp0[0]; bvw0.y = vwp0[bk];
        v2f bvw1; bvw1.x = vwp1[0]; bvw1.y = vwp1[bk];

        // mean: x * W
        cm00 = WMMA_F32(ax0, bw0, cm00);
        cm01 = WMMA_F32(ax0, bw1, cm01);
        cm10 = WMMA_F32(ax1, bw0, cm10);
        cm11 = WMMA_F32(ax1, bw1, cm11);

        // var term: x^2 * vW
        v2f ax20 = ax0 * ax0;
        v2f ax21 = ax1 * ax1;
        cv00 = WMMA_F32(ax20, bvw0, cv00);
        cv01 = WMMA_F32(ax20, bvw1, cv01);
        cv10 = WMMA_F32(ax21, bvw0, cv10);
        cv11 = WMMA_F32(ax21, bvw1, cv11);

        // var term: vx * (W^2 + vW)   (compile-time absent for the embedding)
        if (HAS_VX) {
            v2f avx0; avx0.x = vxp0[0]; avx0.y = vxp0[1];
            v2f avx1; avx1.x = vxp1[0]; avx1.y = vxp1[1];
            v2f bw20 = bw0 * bw0 + bvw0;
            v2f bw21 = bw1 * bw1 + bvw1;
            cv00 = WMMA_F32(avx0, bw20, cv00);
            cv01 = WMMA_F32(avx0, bw21, cv01);
            cv10 = WMMA_F32(avx1, bw20, cv10);
            cv11 = WMMA_F32(avx1, bw21, cv11);
            vxp0 += 4; vxp1 += 4;
        }
        xp0 += 4; xp1 += 4;
        wp0 += 4 * bk; wp1 += 4 * bk;
        vwp0 += 4 * bk; vwp1 += 4 * bk;
    }
#else
    // Scalar fallback (same C fragment layout) — only if the builtin is absent.
    for (int mt = 0; mt < 2; ++mt) {
        for (int i = 0; i < 8; ++i) {
            const int row = tm32 + mt * 16 + i + 8 * half;
            const float* xr2  = xm + xoff + (long long)row * lda;
            const float* vxr2 = HAS_VX ? (xv + xoff + (long long)row * lda) : (const float*)0;
            for (int nt = 0; nt < 2; ++nt) {
                const int ncol = tn32 + nt * 16 + r;
                const float* wr  = wm + woff + (long long)ncol * ldb_n;
                const float* vwr = wv + woff + (long long)ncol * ldb_n;
                float am_ = 0.f, av_ = 0.f;
                for (int k = 0; k < K; ++k) {
                    const float w  = wr[(long long)k * bk];
                    const float vw = vwr[(long long)k * bk];
                    const float xx = xr2[k];
                    am_ += xx * w;
                    av_ += xx * xx * vw + (HAS_VX ? vxr2[k] * (w * w + vw) : 0.f);
                }
                v8f* cmv = (mt == 0) ? (nt == 0 ? &cm00 : &cm01) : (nt == 0 ? &cm10 : &cm11);
                v8f* cvv = (mt == 0) ? (nt == 0 ? &cv00 : &cv01) : (nt == 0 ? &cv10 : &cv11);
                (*cmv)[i] = am_; (*cvv)[i] = av_;
            }
        }
    }
#endif

    // Epilogue. C/D frag: VGPR i holds row (+ i + 8*half), col = tile_n + lane%16.
    const int   col0 = tn32 + r, col1 = col0 + 16;
    const float bm0 = bias  ? bias[col0]  : 0.f;
    const float bm1 = bias  ? bias[col1]  : 0.f;
    const float bv0 = vbias ? vbias[col0] : 0.f;
    const float bv1 = vbias ? vbias[col1] : 0.f;
#pragma unroll
    for (int i = 0; i < 8; ++i) {
        const int row0 = tm32 + i + 8 * half;
        const int row1 = row0 + 16;
        float m00 = cm00[i] * scale_m + bm0;
        float m01 = cm01[i] * scale_m + bm1;
        float m10 = cm10[i] * scale_m + bm0;
        float m11 = cm11[i] * scale_m + bm1;
        float v00 = cv00[i] * scale_v + bv0;
        float v01 = cv01[i] * scale_v + bv1;
        float v10 = cv10[i] * scale_v + bv0;
        float v11 = cv11[i] * scale_v + bv1;
        if (pos) {                                   // x + POS (mean only)
            const long long p0 = (long long)(row0 % posS) * N;
            const long long p1 = (long long)(row1 % posS) * N;
            m00 += pos[p0 + col0]; m01 += pos[p0 + col1];
            m10 += pos[p1 + col0]; m11 += pos[p1 + col1];
        }
        if (do_relu) {                               // relu_vmp
            if (!(m00 > 0.f)) { m00 = 0.f; v00 = 0.f; }
            if (!(m01 > 0.f)) { m01 = 0.f; v01 = 0.f; }
            if (!(m10 > 0.f)) { m10 = 0.f; v10 = 0.f; }
            if (!(m11 > 0.f)) { m11 = 0.f; v11 = 0.f; }
        }
        float* o0 = om + ooff + (long long)row0 * ldo;
        float* o1 = om + ooff + (long long)row1 * ldo;
        float* u0 = ov + ooff + (long long)row0 * ldo;
        float* u1 = ov + ooff + (long long)row1 * ldo;
        o0[col0] = m00; o0[col1] = m01;
        o1[col0] = m10; o1[col1] = m11;
        u0[col0] = v00; u0[col1] = v01;
        u1[col0] = v10; u1[col1] = v11;
    }
}

// ---------------------------------------------------------------------------
// softmax_vmp over last dim: s = softmax(a); vs = (s*(1-s))^2 * va  (in place)
// ---------------------------------------------------------------------------
__global__ __launch_bounds__(256) void softmax_vmp_kernel(
    float* __restrict__ a, float* __restrict__ va, int cols)
{
    const long long row = blockIdx.x;
    float* pa = a  + row * (long long)cols;
    float* pv = va + row * (long long)cols;
    __shared__ float red[256];
    const int tid = threadIdx.x;

    float mx = -3.402823466e38f;
    for (int j = tid; j < cols; j += blockDim.x) mx = fmaxf(mx, pa[j]);
    red[tid] = mx; __syncthreads();
    for (int s = blockDim.x >> 1; s > 0; s >>= 1) {
        if (tid < s) red[tid] = fmaxf(red[tid], red[tid + s]);
        __syncthreads();
    }
    mx = red[0]; __syncthreads();

    float sum = 0.f;
    for (int j = tid; j < cols; j += blockDim.x) {
        const float e = expf(pa[j] - mx);
        pa[j] = e; sum += e;
    }
    red[tid] = sum; __syncthreads();
    for (int s = blockDim.x >> 1; s > 0; s >>= 1) {
        if (tid < s) red[tid] += red[tid + s];
        __syncthreads();
    }
    const float inv = 1.f / red[0];

    for (int j = tid; j < cols; j += blockDim.x) {
        const float sm = pa[j] * inv;
        pa[j] = sm;
        const float t = sm * (1.f - sm);
        pv[j] = t * t * pv[j];
    }
}

// ---------------------------------------------------------------------------
// layernorm_vmp (in place, one block per row)
// ---------------------------------------------------------------------------
__global__ __launch_bounds__(256) void layernorm_vmp_kernel(
    float* __restrict__ x, float* __restrict__ vx,
    const float* __restrict__ g,  const float* __restrict__ b,
    const float* __restrict__ vg, const float* __restrict__ vb, int Dd)
{
    const long long row = blockIdx.x;
    float* px = x  + row * (long long)Dd;
    float* pv = vx + row * (long long)Dd;
    __shared__ float r1[256], r2[256];
    const int tid = threadIdx.x;

    float s = 0.f, s2 = 0.f;
    for (int j = tid; j < Dd; j += blockDim.x) { const float t = px[j]; s += t; s2 += t * t; }
    r1[tid] = s; r2[tid] = s2; __syncthreads();
    for (int st = blockDim.x >> 1; st > 0; st >>= 1) {
        if (tid < st) { r1[tid] += r1[tid + st]; r2[tid] += r2[tid + st]; }
        __syncthreads();
    }
    const float mu   = r1[0] / (float)Dd;
    const float var  = r2[0] / (float)Dd - mu * mu;   // population variance (ddof=0)
    const float inv  = 1.f / (var + LN_TOL);
    const float rstd = sqrtf(inv);

    for (int j = tid; j < Dd; j += blockDim.x) {
        const float xn  = (px[j] - mu) * rstd;
        const float vxn = pv[j] * inv;
        const float gg = g[j], vgg = vg[j];
        px[j] = gg * xn + b[j];
        pv[j] = (gg * gg + vgg) * vxn + xn * xn * vgg + vb[j];
    }
}

__global__ void residual_add_kernel(float* __restrict__ xm, float* __restrict__ xv,
                                    const float* __restrict__ om,
                                    const float* __restrict__ ov, long long n)
{
    const long long i = (long long)blockIdx.x * blockDim.x + threadIdx.x;
    if (i < n) { xm[i] += om[i]; xv[i] += ov[i]; }
}

__global__ void pos_init_kernel(float* __restrict__ pos)
{
    const long long i = (long long)blockIdx.x * blockDim.x + threadIdx.x;
    if (i >= (long long)S_LEN * DMODEL) return;
    const int t = (int)(i / DMODEL), d = (int)(i % DMODEL);
    const double angle = (double)t / pow(10000.0, (double)(2 * (d / 2)) / (double)DMODEL);
    pos[i] = (d % 2 == 0) ? (float)sin(angle) : (float)cos(angle);
}

// Output: split_heads(k), split_heads(vk), split_heads(v), split_heads(vv)
__global__ void split_heads_out_kernel(const float* __restrict__ km, const float* __restrict__ kv,
                                       const float* __restrict__ vm, const float* __restrict__ vv,
                                       float* __restrict__ out)
{
    const long long C = (long long)NBATCH * NHEAD * S_LEN * DHEAD;
    const long long i = (long long)blockIdx.x * blockDim.x + threadIdx.x;
    if (i >= C) return;
    const int d = (int)(i % DHEAD);
    long long t = i / DHEAD;
    const int s = (int)(t % S_LEN); t /= S_LEN;
    const int h = (int)(t % NHEAD);
    const int b = (int)(t / NHEAD);
    const long long src = ((long long)(b * S_LEN + s)) * DMODEL + h * DHEAD + d;
    out[i]         = km[src];
    out[C + i]     = kv[src];
    out[2 * C + i] = vm[src];
    out[3 * C + i] = vv[src];
}

// ---------------------------------------------------------------------------
extern "C" void kernel_launch(void* const* d_in, const int* in_sizes, int n_in,
                              void* d_out, int out_size, void* d_ws, size_t ws_size,
                              hipStream_t stream)
{
    (void)in_sizes; (void)n_in; (void)out_size;

    const float* x      = (const float*)d_in[0];
    const float* emb_W  = (const float*)d_in[1];
    const float* emb_vW = (const float*)d_in[2];
    const float* emb_b  = (const float*)d_in[3];
    const float* emb_vb = (const float*)d_in[4];
    // layer i, tensor j: Wq,vWq,Wk,vWk,Wv,vWv,fc_W,fc_vW,fc_b,fc_vb,n2_g,n2_b,n2_vg,n2_vb
    #define LP(i, j)  ((const float*)d_in[5  + (i) * 14 + (j)])
    #define NP(i, j)  ((const float*)d_in[33 + (i) * 4  + (j)])  // g,b,vg,vb
    const float* fin_Wk  = (const float*)d_in[41];
    const float* fin_vWk = (const float*)d_in[42];
    const float* fin_Wv  = (const float*)d_in[43];
    const float* fin_vWv = (const float*)d_in[44];

    // ---- workspace layout (floats) ----
    float* ws = (float*)d_ws;
    size_t o = 0;
    float* POS = ws + o; o += (size_t)S_LEN * DMODEL;
    float* xm  = ws + o; o += (size_t)BS * DMODEL;
    float* xv  = ws + o; o += (size_t)BS * DMODEL;
    float* qm  = ws + o; o += (size_t)BS * DMODEL;
    float* qv  = ws + o; o += (size_t)BS * DMODEL;
    float* kkm = ws + o; o += (size_t)BS * DMODEL;
    float* kkv = ws + o; o += (size_t)BS * DMODEL;
    float* vvm = ws + o; o += (size_t)BS * DMODEL;
    float* vvv = ws + o; o += (size_t)BS * DMODEL;
    float* om  = ws + o; o += (size_t)BS * DMODEL;
    float* ov  = ws + o; o += (size_t)BS * DMODEL;
    // attention score chunk buffers: sized so the score working set stays
    // inside the 192MB L2 (<= 8 batches * 2 tensors * 4MB = 64MB).
    const size_t perBH = 2ull * S_LEN * S_LEN;           // am+av floats per (b,h)
    size_t remFloats = (ws_size / 4 > o) ? (ws_size / 4 - o) : 0;
    int CH = (int)(remFloats / perBH);
    if (CH > 8)  CH = 8;
    if (CH > BH) CH = BH;
    if (CH < 1)  CH = 1;
    float* am = ws + o;
    float* av = am + (size_t)CH * S_LEN * S_LEN;

    const float RD = sqrtf((float)DMODEL);

    // ---- positional encoding ----
    {
        const long long n = (long long)S_LEN * DMODEL;
        pos_init_kernel<<<(unsigned)((n + 255) / 256), 256, 0, stream>>>(POS);
    }

    // Plain [BS,DMODEL] <- [BS,K] x [DMODEL,K]^T linear_vmp  (supertiles % 8 == 0)
    auto linear = [&](const float* Xm, const float* Xv,
                      const float* Wm, const float* Wv,
                      const float* bb, const float* vb2, const float* pos,
                      float* Om, float* Ov, int K, int relu) {
        const long long st = (long long)(BS / 32) * (DMODEL / 32);     // 2048
        dim3 grid((unsigned)(st / 8), 1);
        if (Xv) {
            gemm_vmp_wmma<true, true><<<grid, 256, 0, stream>>>(
                Xm, Xv, Wm, Wv, bb, vb2, pos, S_LEN,
                Om, Ov, DMODEL, K,
                /*lda*/K, /*ldb_n*/(long long)K, /*ldb_k*/1LL, /*ldo*/DMODEL,
                /*Hdim*/1, 0, 0, 0, 0, 0, 0, 0, 0, 0,
                1.f, 1.f, relu);
        } else {
            gemm_vmp_wmma<false, true><<<grid, 256, 0, stream>>>(
                Xm, Xv, Wm, Wv, bb, vb2, pos, S_LEN,
                Om, Ov, DMODEL, K,
                /*lda*/K, /*ldb_n*/(long long)K, /*ldb_k*/1LL, /*ldo*/DMODEL,
                /*Hdim*/1, 0, 0, 0, 0, 0, 0, 0, 0, 0,
                1.f, 1.f, relu);
        }
    };

    auto layernorm = [&](const float* g, const float* b,
                         const float* vg, const float* vb) {
        layernorm_vmp_kernel<<<BS, 256, 0, stream>>>(xm, xv, g, b, vg, vb, DMODEL);
    };

    // ---- embedding: vx = 0 -> variance = x^2 @ emb_vW^T + emb_vb; mean += POS ----
    linear(x, (const float*)0, emb_W, emb_vW, emb_b, emb_vb, POS, xm, xv, DIN, 0);

    // ---- 2 encoder layers ----
    for (int i = 0; i < 2; ++i) {
        if (i != 0) layernorm(NP(i - 1, 0), NP(i - 1, 1), NP(i - 1, 2), NP(i - 1, 3));

        linear(xm, xv, LP(i, 0), LP(i, 1), 0, 0, 0, qm,  qv,  DMODEL, 0);  // Q
        linear(xm, xv, LP(i, 2), LP(i, 3), 0, 0, 0, kkm, kkv, DMODEL, 0);  // K
        linear(xm, xv, LP(i, 4), LP(i, 5), 0, 0, 0, vvm, vvv, DMODEL, 0);  // V

        // attention, chunked over (b,h) so scores stay in L2
        for (int bh0 = 0; bh0 < BH; bh0 += CH) {
            const int nb = (BH - bh0 < CH) ? (BH - bh0) : CH;

            { // scores: a = q k^T / sqrt(D);  va = (vq(k^2+vk) + q^2 vk) / D
                const long long st = (long long)(S_LEN / 32) * (S_LEN / 32); // 1024
                dim3 grid((unsigned)(st / 8), (unsigned)nb);
                gemm_vmp_wmma<true, true><<<grid, 256, 0, stream>>>(
                    qm, qv, kkm, kkv, 0, 0, 0, 1,
                    am, av, S_LEN, DHEAD,
                    /*lda*/DMODEL, /*ldb_n*/(long long)DMODEL, /*ldb_k*/1LL, /*ldo*/S_LEN,
                    NHEAD,
                    bh0, (long long)S_LEN * DMODEL, (long long)DHEAD,
                    bh0, (long long)S_LEN * DMODEL, (long long)DHEAD,
                    0,   (long long)NHEAD * S_LEN * S_LEN, (long long)S_LEN * S_LEN,
                    1.f / RD, 1.f / (RD * RD), 0);
            }

            softmax_vmp_kernel<<<(unsigned)(nb * S_LEN), 256, 0, stream>>>(am, av, S_LEN);

            { // out: o = s v ; vo = vs(v^2+vv) + s^2 vv    (B strided: ldb_k = D)
                const long long st = (long long)(S_LEN / 32) * (DHEAD / 32);  // 64
                dim3 grid((unsigned)(st / 8), (unsigned)nb);
                gemm_vmp_wmma<true, false><<<grid, 256, 0, stream>>>(
                    am, av, vvm, vvv, 0, 0, 0, 1,
                    om, ov, DHEAD, S_LEN,
                    /*lda*/S_LEN, /*ldb_n*/1LL, /*ldb_k*/(long long)DMODEL, /*ldo*/DMODEL,
                    NHEAD,
                    0,   (long long)NHEAD * S_LEN * S_LEN, (long long)S_LEN * S_LEN,
                    bh0, (long long)S_LEN * DMODEL, (long long)DHEAD,
                    bh0, (long long)S_LEN * DMODEL, (long long)DHEAD,
                    1.f, 1.f, 0);
            }
        }

        const long long n = (long long)BS * DMODEL;
        residual_add_kernel<<<(unsigned)((n + 255) / 256), 256, 0, stream>>>(xm, xv, om, ov, n);
        layernorm(LP(i, 10), LP(i, 11), LP(i, 12), LP(i, 13));          // n2
        // x0 = x; fc + relu into om/ov; residual
        linear(xm, xv, LP(i, 6), LP(i, 7), LP(i, 8), LP(i, 9), 0, om, ov, DMODEL, 1);
        residual_add_kernel<<<(unsigned)((n + 255) / 256), 256, 0, stream>>>(xm, xv, om, ov, n);
    }

    // ---- final norm + projections + split heads ----
    layernorm(NP(1, 0), NP(1, 1), NP(1, 2), NP(1, 3));
    linear(xm, xv, fin_Wk, fin_vWk, 0, 0, 0, kkm, kkv, DMODEL, 0);
    linear(xm, xv, fin_Wv, fin_vWv, 0, 0, 0, vvm, vvv, DMODEL, 0);

    {
        const long long C = (long long)NBATCH * NHEAD * S_LEN * DHEAD;
        split_heads_out_kernel<<<(unsigned)((C + 255) / 256), 256, 0, stream>>>(
            kkm, kkv, vvm, vvv, (float*)d_out);
    }
    #undef LP
    #undef NP
}